// LinearErnie4_5_MoeSparseMoeBlock_40458591928882
// MI455X (gfx1250) — compile-verified
//
#include <hip/hip_runtime.h>
#include <hip/hip_bf16.h>

// ---------------- problem constants ----------------
#define TT   4096          // B*S tokens
#define HD   2560          // hidden
#define NE   32            // experts
#define KTOP 6             // top-k
#define FF   1536          // expert intermediate
#define SFF  3072          // shared intermediate
#define TOTROWS (TT*KTOP)  // 24576 routed rows (exact, routing-independent)

typedef float v2f  __attribute__((ext_vector_type(2)));
typedef float v8f  __attribute__((ext_vector_type(8)));
typedef unsigned int u32x4 __attribute__((ext_vector_type(4)));
typedef int   i32x4 __attribute__((ext_vector_type(4)));
typedef int   i32x8 __attribute__((ext_vector_type(8)));

// ---------------- router: 1 wave per token ----------------
__global__ __launch_bounds__(32)
void router_kernel(const float* __restrict__ X, const float* __restrict__ RW,
                   const float* __restrict__ RB,
                   int* __restrict__ top_i, float* __restrict__ top_w,
                   int* __restrict__ counts) {
  const int t = blockIdx.x;
  const int lane = threadIdx.x;                  // lane == expert id
  const float4* xp = (const float4*)(X + (long long)t * HD);
  const float4* wp = (const float4*)(RW + (long long)lane * HD);
  float acc = 0.f;
  for (int i = 0; i < HD / 4; ++i) {
    float4 a = xp[i], b = wp[i];
    acc += a.x * b.x + a.y * b.y + a.z * b.z + a.w * b.w;
  }
  // softmax over 32 lanes
  float m = acc;
  for (int off = 16; off; off >>= 1) m = fmaxf(m, __shfl_xor(m, off, 32));
  float ex = __expf(acc - m);
  float s = ex;
  for (int off = 16; off; off >>= 1) s += __shfl_xor(s, off, 32);
  float score = ex / s + RB[lane];
  // top-6 via repeated argmax reduction
  float wk[KTOP]; int ik[KTOP];
  float wsum = 0.f;
  for (int k = 0; k < KTOP; ++k) {
    float v = score; int idx = lane;
    for (int off = 16; off; off >>= 1) {
      float ov = __shfl_xor(v, off, 32);
      int   oi = __shfl_xor(idx, off, 32);
      if (ov > v || (ov == v && oi < idx)) { v = ov; idx = oi; }
    }
    wk[k] = v; ik[k] = idx; wsum += v;
    if (lane == idx) score = -__builtin_inff();
  }
  float inv = 1.f / fmaxf(wsum, 1e-12f);
  if (lane == 0) {
    for (int k = 0; k < KTOP; ++k) {
      top_i[t * KTOP + k] = ik[k];
      top_w[t * KTOP + k] = wk[k] * inv;
      atomicAdd(&counts[ik[k]], 1);
    }
  }
}

// ---------------- exclusive scan of 32 counts ----------------
__global__ void scan_kernel(const int* __restrict__ counts, int* __restrict__ offsets) {
  if (threadIdx.x == 0) {
    int acc = 0;
    for (int e = 0; e < NE; ++e) { offsets[e] = acc; acc += counts[e]; }
    offsets[NE] = acc;
  }
}

// ---------------- scatter token->expert segments ----------------
__global__ __launch_bounds__(256)
void scatter_kernel(const int* __restrict__ top_i, const float* __restrict__ top_w,
                    const int* __restrict__ offsets, int* __restrict__ cursor,
                    int* __restrict__ perm, float* __restrict__ pw) {
  int t = blockIdx.x * blockDim.x + threadIdx.x;
  if (t >= TT) return;
  for (int k = 0; k < KTOP; ++k) {
    int e = top_i[t * KTOP + k];
    int pos = offsets[e] + atomicAdd(&cursor[e], 1);
    perm[pos] = t;
    pw[pos] = top_w[t * KTOP + k];
  }
}

// ---- TDM: async-load one 16x128 fp32 tile into LDS, K-PAIR INTERLEAVED ----
// 4-D transfer: X=1 elem, Y=2 rows (stride ldb), Z=128 cols (stride 1),
// ZZ=8 row-pairs (stride 2*ldb)  ==>  LDS stream is (B[k][n],B[k+1][n]) pairs,
// i.e. exactly the f32-WMMA B-fragment register layout.
// pad_interval=7 (every 256 DWORDs = one pair-row) + pad_amount=31 (32 DWORDs)
// gives a 288-DWORD pair-row stride (bank-conflict-free half-wave b64 reads).
#define BPR 288   // floats per pair-row in LDS (256 data + 32 pad)

__device__ __forceinline__ void tdm_load_tile(const float* gptr, unsigned lds_off,
                                              int ld_elems) {
  unsigned long long ga = (unsigned long long)(const void*)gptr;
  u32x4 g0;
  g0[0] = 1u;                                            // count=1, user mode
  g0[1] = lds_off;                                       // LDS byte address
  g0[2] = (unsigned)ga;                                  // global addr [31:0]
  g0[3] = (unsigned)((ga >> 32) & 0x01FFFFFFull) | (2u << 30);  // [56:32] | type=2
  unsigned ld = (unsigned)ld_elems;
  i32x8 g1;
  g1[0] = (int)((2u << 16) | (1u << 20) | (7u << 22) | (31u << 25)); // 4B, pad 256+32 DW
  g1[1] = (int)(1u << 16);               // tensor_dim0 = 1 (lo16; atomic addr = 0)
  g1[2] = (int)(2u << 16);               // tensor_dim0 hi=0 | tensor_dim1 = 2
  g1[3] = (int)(1u << 16);               // tensor_dim1 hi=0 | tile_dim0 = 1
  g1[4] = (int)(2u | (128u << 16));      // tile_dim1 = 2 rows | tile_dim2 = 128 cols
  g1[5] = (int)ld;                       // tensor_dim0_stride = ldb (lo32)
  g1[6] = (int)(1u << 16);               // stride0 hi=0 | tensor_dim1_stride = 1 (lo16)
  g1[7] = 0;                             // tensor_dim1_stride hi = 0
  i32x4 g2;
  g2[0] = 128;                           // tensor_dim2 = 128
  g2[1] = 8;                             // tensor_dim3 = 8
  g2[2] = (int)(2u * ld);                // tensor_dim2_stride = 2*ldb (lo32)
  g2[3] = (int)(8u << 16);               // stride2 hi=0 | tile_dim3 = 8 row-pairs
  i32x4 z4 = {0, 0, 0, 0};
#if defined(__clang_major__) && (__clang_major__ >= 23)
  i32x8 z8 = {0, 0, 0, 0, 0, 0, 0, 0};
  __builtin_amdgcn_tensor_load_to_lds(g0, g1, g2, z4, z8, 0);
#else
  __builtin_amdgcn_tensor_load_to_lds(g0, g1, g2, z4, 0);
#endif
}

// ---------------- fused tiled GEMM with WMMA f32 16x16x4 + TDM pipeline ----------
// Tile: 128 (M) x 128 (N), K-step 16. 512 threads = 16 waves (4x4 wave grid),
// each wave owns a 32x32 output block (2x2 WMMA tiles).
// B tiles: double-buffered LDS (pair-interleaved), fed by Tensor Data Mover.
// A tiles: double-buffered LDS, software-pipelined VGPR staging.
// GATEUP:  D[row, n] = silu(A@B0) * (A@B1)          (writes ACT)
// !GATEUP: atomicAdd(OUT[row, n], scale * (A@B0))   (down-proj combine)
#define TM 128
#define TN 128
#define TK 16
#define TKP 20        // padded LDS stride for A (bank-conflict free)

template<bool GATEUP, bool ROUTED>
__global__ __launch_bounds__(512)
void gemm_kernel(const float* __restrict__ A,
                 const float* __restrict__ B0,
                 const float* __restrict__ B1,
                 float* __restrict__ D,
                 const int* __restrict__ offsets,
                 const int* __restrict__ perm,
                 const float* __restrict__ pw,
                 int Mfixed, int Kdim,
                 int lda, int ldb, int ldd,
                 long long strideB) {
  __shared__ float sA[2][TM][TKP];
  __shared__ float sB0[2][8 * BPR];                 // 8 pair-rows x 288 floats
  __shared__ float sB1[GATEUP ? 2 : 1][8 * BPR];
  constexpr int NB = GATEUP ? 2 : 1;      // tensor ops per K-tile

  const int e = ROUTED ? blockIdx.z : 0;
  int base = 0, mlen = Mfixed;
  if (ROUTED) { base = offsets[e]; mlen = offsets[e + 1] - base; }
  const int tile_m = blockIdx.y * TM;
  if (tile_m >= mlen) return;                      // empty / out-of-segment tile
  const int n0 = blockIdx.x * TN;

  const float* Bp0 = B0 + (long long)e * strideB;
  const float* Bp1 = GATEUP ? (B1 + (long long)e * strideB) : B0;

  const int tid  = threadIdx.x;
  const int lane = tid & 31;
  const int wave = tid >> 5;
  const int wr = wave & 3;        // 32-row block
  const int wc = wave >> 2;       // 32-col block
  const int half = lane >> 4;     // 0: lanes 0-15, 1: lanes 16-31
  const int ml = lane & 15;

  v8f acc0[2][2];
  v8f acc1[GATEUP ? 2 : 1][2];
  const v8f vzero = {0.f,0.f,0.f,0.f,0.f,0.f,0.f,0.f};
#pragma unroll
  for (int ri = 0; ri < 2; ++ri)
#pragma unroll
    for (int ji = 0; ji < 2; ++ji) {
      acc0[ri][ji] = vzero;
      if constexpr (GATEUP) acc1[ri][ji] = vzero;
    }

  // A-load mapping: 512 threads -> 128 rows x 16 cols (1 float4 each)
  const int la_r = tid >> 2;
  const int la_k = (tid & 3) * 4;
  const bool a_valid = (tile_m + la_r) < mlen;
  long long a_row;
  if (ROUTED) {
    if (GATEUP) a_row = a_valid ? (long long)perm[base + tile_m + la_r] : 0ll;
    else        a_row = a_valid ? (long long)(base + tile_m + la_r) : (long long)base;
  } else {
    a_row = a_valid ? (long long)(tile_m + la_r) : 0ll;
  }
  const float* Arow = A + a_row * lda;

  const int ntiles = Kdim / TK;

  // ---- pipeline prologue: TDM tiles 0,1 in flight; A(0) staged to LDS ----
  if (wave == 0) {
    tdm_load_tile(Bp0 + n0, (unsigned)(unsigned long long)(void*)&sB0[0][0], ldb);
    if constexpr (GATEUP)
      tdm_load_tile(Bp1 + n0, (unsigned)(unsigned long long)(void*)&sB1[0][0], ldb);
    if (ntiles > 1) {
      tdm_load_tile(Bp0 + (long long)TK * ldb + n0,
                    (unsigned)(unsigned long long)(void*)&sB0[1][0], ldb);
      if constexpr (GATEUP)
        tdm_load_tile(Bp1 + (long long)TK * ldb + n0,
                      (unsigned)(unsigned long long)(void*)&sB1[1][0], ldb);
    }
  }
  {
    float4 av = make_float4(0.f, 0.f, 0.f, 0.f);
    if (a_valid) av = *(const float4*)(Arow + la_k);
    *(float4*)&sA[0][la_r][la_k] = av;
  }

  for (int i = 0; i < ntiles; ++i) {
    const int p = i & 1;
    // stage A(i+1) into registers early (overlaps with tensor wait + barrier)
    float4 av_n = make_float4(0.f, 0.f, 0.f, 0.f);
    if ((i + 1) < ntiles && a_valid)
      av_n = *(const float4*)(Arow + (long long)(i + 1) * TK + la_k);

    // wait until tile i's TDM writes landed (tile i+1's NB ops may stay in flight)
    if ((i + 1) < ntiles) __builtin_amdgcn_s_wait_tensorcnt(NB);
    else                  __builtin_amdgcn_s_wait_tensorcnt(0);
    __syncthreads();                               // B-tile i visible; prev compute done

    if ((i + 1) < ntiles)
      *(float4*)&sA[1 - p][la_r][la_k] = av_n;     // stage A(i+1)

#pragma unroll
    for (int kk = 0; kk < 4; ++kk) {
      const int kc = kk * 4 + 2 * half;            // this half-wave's K columns
      const int pidx = kk * 2 + half;              // pair-row index in LDS B tiles
      v2f a[2];
#pragma unroll
      for (int ri = 0; ri < 2; ++ri) {
        const int mrow = wr * 32 + ri * 16 + ml;
        a[ri].x = sA[p][mrow][kc];
        a[ri].y = sA[p][mrow][kc + 1];
      }
#pragma unroll
      for (int ji = 0; ji < 2; ++ji) {
        const int n = wc * 32 + ji * 16 + ml;
        v2f b0v = *(const v2f*)&sB0[p][pidx * BPR + 2 * n];   // single b64 load
        v2f b1v;
        if constexpr (GATEUP) b1v = *(const v2f*)&sB1[p][pidx * BPR + 2 * n];
#pragma unroll
        for (int ri = 0; ri < 2; ++ri) {
          acc0[ri][ji] = __builtin_amdgcn_wmma_f32_16x16x4_f32(
              false, a[ri], false, b0v, (short)0, acc0[ri][ji], false, false);
          if constexpr (GATEUP)
            acc1[ri][ji] = __builtin_amdgcn_wmma_f32_16x16x4_f32(
                false, a[ri], false, b1v, (short)0, acc1[ri][ji], false, false);
        }
      }
    }
    __syncthreads();                               // all waves done reading buffer p

    if (wave == 0 && (i + 2) < ntiles) {           // refill buffer p with tile i+2
      tdm_load_tile(Bp0 + (long long)(i + 2) * TK * ldb + n0,
                    (unsigned)(unsigned long long)(void*)&sB0[p][0], ldb);
      if constexpr (GATEUP)
        tdm_load_tile(Bp1 + (long long)(i + 2) * TK * ldb + n0,
                      (unsigned)(unsigned long long)(void*)&sB1[p][0], ldb);
    }
  }

  // Epilogue. C layout: VGPR r -> M = r (lanes 0-15) / r+8 (lanes 16-31), N = lane&15.
#pragma unroll
  for (int ri = 0; ri < 2; ++ri) {
#pragma unroll
    for (int ji = 0; ji < 2; ++ji) {
      const int n = n0 + wc * 32 + ji * 16 + ml;
#pragma unroll
      for (int r = 0; r < 8; ++r) {
        const int m_local = wr * 32 + ri * 16 + half * 8 + r;
        const int rm = tile_m + m_local;
        if (rm < mlen) {
          if constexpr (GATEUP) {
            float g = acc0[ri][ji][r];
            float u = acc1[ri][ji][r];
            float act = g / (1.f + __expf(-g)) * u;   // silu(g) * u
            long long drow = ROUTED ? (long long)(base + rm) : (long long)rm;
            D[drow * ldd + n] = act;
          } else {
            float v = acc0[ri][ji][r];
            long long orow;
            if (ROUTED) { v *= pw[base + rm]; orow = (long long)perm[base + rm]; }
            else        { orow = (long long)rm; }
            atomicAdd(&D[orow * ldd + n], v);
          }
        }
      }
    }
  }
}

// ---------------- host launcher ----------------
extern "C" void kernel_launch(void* const* d_in, const int* in_sizes, int n_in,
                              void* d_out, int out_size, void* d_ws, size_t ws_size,
                              hipStream_t stream) {
  const float* X   = (const float*)d_in[0];   // [T, H]
  const float* RW  = (const float*)d_in[1];   // [E, H]
  const float* RB  = (const float*)d_in[2];   // [E]
  const float* GW  = (const float*)d_in[3];   // [E, H, F]
  const float* UW  = (const float*)d_in[4];   // [E, H, F]
  const float* DW  = (const float*)d_in[5];   // [E, F, H]
  const float* SGW = (const float*)d_in[6];   // [H, SF]
  const float* SUW = (const float*)d_in[7];   // [H, SF]
  const float* SDW = (const float*)d_in[8];   // [SF, H]
  float* OUT = (float*)d_out;                 // [T, H]

  char* ws = (char*)d_ws;
  int*   counts  = (int*)(ws + 0);                       // 32 ints
  int*   cursor  = (int*)(ws + 128);                     // 32 ints
  int*   offsets = (int*)(ws + 256);                     // 33 ints
  int*   top_i   = (int*)(ws + 512);                     // T*K
  float* top_w   = (float*)(ws + 512 + (size_t)TOTROWS * 4);
  int*   perm    = (int*)(ws + 512 + (size_t)TOTROWS * 8);
  float* pw      = (float*)(ws + 512 + (size_t)TOTROWS * 12);
  // ACT region reused: shared phase (T*SF floats) then routed phase (T*K*F floats)
  float* ACT     = (float*)(ws + 512 + (size_t)TOTROWS * 16);

  hipMemsetAsync(ws, 0, 512, stream);                          // counts+cursor+offsets
  hipMemsetAsync(OUT, 0, (size_t)TT * HD * sizeof(float), stream);

  router_kernel<<<TT, 32, 0, stream>>>(X, RW, RB, top_i, top_w, counts);
  scan_kernel<<<1, 32, 0, stream>>>(counts, offsets);
  scatter_kernel<<<TT / 256, 256, 0, stream>>>(top_i, top_w, offsets, cursor, perm, pw);

  // ---- shared experts (dense) ----
  gemm_kernel<true, false><<<dim3(SFF / TN, TT / TM), 512, 0, stream>>>(
      X, SGW, SUW, ACT, nullptr, nullptr, nullptr,
      TT, HD, HD, SFF, SFF, 0ll);
  gemm_kernel<false, false><<<dim3(HD / TN, TT / TM), 512, 0, stream>>>(
      ACT, SDW, nullptr, OUT, nullptr, nullptr, nullptr,
      TT, SFF, SFF, HD, HD, 0ll);

  // ---- routed experts (grouped over segments) ----
  gemm_kernel<true, true><<<dim3(FF / TN, TT / TM, NE), 512, 0, stream>>>(
      X, GW, UW, ACT, offsets, perm, pw,
      0, HD, HD, FF, FF, (long long)HD * FF);
  gemm_kernel<false, true><<<dim3(HD / TN, TT / TM, NE), 512, 0, stream>>>(
      ACT, DW, nullptr, OUT, offsets, perm, pw,
      0, FF, FF, HD, HD, (long long)FF * HD);
}